// PathEmbeding2D_25744033972437
// MI455X (gfx1250) — compile-verified
//
#include <hip/hip_runtime.h>
#include <cstdint>

typedef __attribute__((ext_vector_type(2))) float v2f;
typedef __attribute__((ext_vector_type(4))) float v4f;
typedef __attribute__((ext_vector_type(8))) float v8f;

#define HH 56
#define WW 56
#define NH 8
#define HD 32
#define SEQ (HH * WW)

__device__ __forceinline__ float softplus_f(float z) {
    // stable softplus: max(z,0) + log1p(exp(-|z|))
    return fmaxf(z, 0.0f) + log1pf(expf(-fabsf(z)));
}

// ---------------------------------------------------------------------------
// Kernel 1a: dt projection + softplus + scale by A = -exp(A_log).
// One thread per (h, w, n). dtA/dtB layout: [(h*56+w)*8 + n].
// ---------------------------------------------------------------------------
__global__ void dt_kernel(const float* __restrict__ x, const float* __restrict__ Wdt,
                          const float* __restrict__ bias, const float* __restrict__ Alog,
                          float* __restrict__ dtA, float* __restrict__ dtB) {
    int tid = blockIdx.x * blockDim.x + threadIdx.x;
    if (tid >= SEQ * NH) return;
    int n  = tid & 7;
    int hw = tid >> 3;
    const float* xb = x + (size_t)hw * (NH * HD) + n * HD;
    float d0 = 0.f, d1 = 0.f;
#pragma unroll
    for (int d = 0; d < HD; ++d) {
        float xv = xb[d];
        d0 += xv * Wdt[d];
        d1 += xv * Wdt[HD + d];
    }
    float A = -expf(Alog[n]);
    float b = bias[n];
    dtA[tid] = softplus_f(d0 + b) * A;
    dtB[tid] = softplus_f(d1 + b) * A;
}

// ---------------------------------------------------------------------------
// Kernel 1b: cumsum via triangular-ones matmul on the f32 WMMA path.
//   C(64x64) = A(64x64) @ T(64x64), T[k][j] = (k <= j).
// One block per (head n, direction). 16 waves; each wave owns one 16x16
// output tile (mi, nj) and chains 16 V_WMMA_F32_16X16X4_F32 ops over K=64.
// A-frag (16x4 f32): lane<16 -> K={kb,kb+1}; lane>=16 -> K={kb+2,kb+3}.
// C/D frag: lane N = lane&15; VGPR r -> M = r + 8*(lane>=16).
// cw layout: [(h*8+n)*56 + w]; ch layout: [(w*8+n)*56 + h].
// ---------------------------------------------------------------------------
__global__ void cumsum_wmma_kernel(const float* __restrict__ dtA, const float* __restrict__ dtB,
                                   float* __restrict__ cw, float* __restrict__ ch) {
    const int n   = blockIdx.x >> 1;
    const int dir = blockIdx.x & 1;
    const float* src = dir ? dtB : dtA;
    float* dst       = dir ? ch  : cw;

    __shared__ float smat[64 * 65];   // padded stride: conflict-free column access
    const int tid = threadIdx.x;
    for (int i = tid; i < 64 * 64; i += 512) {
        int r = i >> 6, c = i & 63;
        float v = 0.f;
        if (r < HH && c < WW) {
            // dir 0: row=h, col=w, src = dtA[(h*56+w)*8+n]
            // dir 1: row=w, col=h, src = dtB[(h*56+w)*8+n] with h=c, w=r
            v = dir ? src[(c * WW + r) * NH + n] : src[(r * WW + c) * NH + n];
        }
        smat[r * 65 + c] = v;
    }
    __syncthreads();

    const int wave    = tid >> 5;
    const int lane    = tid & 31;
    const int mi      = wave >> 2;        // 0..3 output row tile
    const int nj      = wave & 3;         // 0..3 output col tile
    const int halfsel = (lane >= 16) ? 1 : 0;
    const int l15     = lane & 15;
    const int row     = mi * 16 + l15;    // A-matrix row for this lane
    const int col     = nj * 16 + l15;    // C/D + B column for this lane

    v8f acc = {};
#pragma unroll
    for (int kt = 0; kt < 16; ++kt) {
        int kb = kt * 4 + 2 * halfsel;    // global K index pair base
        v2f a, b;
        a.x = smat[row * 65 + kb];
        a.y = smat[row * 65 + kb + 1];
        b.x = (kb     <= col) ? 1.0f : 0.0f;   // T[k][j] = (k <= j)
        b.y = (kb + 1 <= col) ? 1.0f : 0.0f;
        acc = __builtin_amdgcn_wmma_f32_16x16x4_f32(
            /*neg_a=*/false, a, /*neg_b=*/false, b,
            /*c_mod=*/(short)0, acc, /*reuse_a=*/false, /*reuse_b=*/false);
    }

#pragma unroll
    for (int r = 0; r < 8; ++r) {
        int grow = mi * 16 + r + 8 * halfsel;
        if (grow < HH && col < WW)
            dst[(grow * NH + n) * WW + col] = acc[r];
    }
}

// ---------------------------------------------------------------------------
// Kernel 2: bandwidth-bound writer. One block per (n, h1, h2) 56x56 tile.
// tile[w1][w2] = segsum_w(crow)[w1,w2] + (ch[w2,n,max(h1,h2)] - ch[w2,n,min]).
// 14 float4 NT stores per row; all 16B-aligned (3136*4 and 224 are mult of 16).
// ---------------------------------------------------------------------------
__global__ void mask_write_kernel(const float* __restrict__ cw, const float* __restrict__ ch,
                                  float* __restrict__ out) {
    int bx  = blockIdx.x;
    int n   = bx / SEQ;
    int rem = bx - n * SEQ;
    int h1  = rem / WW;
    int h2  = rem - h1 * WW;

    __shared__ float crow[WW];
    __shared__ float vcol[WW];
    int tid = threadIdx.x;
    int hmn = (h1 < h2) ? h1 : h2;
    int hmx = (h1 < h2) ? h2 : h1;
    if (tid < WW) {
        crow[tid] = cw[(h1 * NH + n) * WW + tid];
        const float* chp = ch + (tid * NH + n) * HH;
        vcol[tid] = chp[hmx] - chp[hmn];
    }
    __syncthreads();

    size_t base = ((size_t)n * SEQ + (size_t)h1 * WW) * SEQ + (size_t)h2 * WW;
    for (int idx = tid; idx < WW * 14; idx += 256) {
        int w1 = idx / 14;
        int q  = idx - w1 * 14;
        int c0 = q * 4;
        float a1 = crow[w1];
        v4f r;
#pragma unroll
        for (int j = 0; j < 4; ++j) {
            int w2   = c0 + j;
            float c2 = crow[w2];
            float sw = (w1 >= w2) ? (a1 - c2) : (c2 - a1);
            r[j] = sw + vcol[w2];
        }
        __builtin_nontemporal_store(r, (v4f*)(out + base + (size_t)w1 * SEQ + c0));
    }
}

// ---------------------------------------------------------------------------
extern "C" void kernel_launch(void* const* d_in, const int* in_sizes, int n_in,
                              void* d_out, int out_size, void* d_ws, size_t ws_size,
                              hipStream_t stream) {
    (void)in_sizes; (void)n_in; (void)out_size; (void)ws_size;
    const float* x    = (const float*)d_in[0];   // (1,56,56,256)
    const float* Wdt  = (const float*)d_in[1];   // (2,32)
    const float* bias = (const float*)d_in[2];   // (8,)
    const float* Alog = (const float*)d_in[3];   // (8,)
    float* out = (float*)d_out;

    float* ws  = (float*)d_ws;                   // 4 * 25088 floats = ~400 KB
    const int NELEM = SEQ * NH;                  // 25088
    float* dtA = ws;
    float* dtB = ws + NELEM;
    float* cw  = ws + 2 * NELEM;
    float* ch  = ws + 3 * NELEM;

    dt_kernel<<<(NELEM + 255) / 256, 256, 0, stream>>>(x, Wdt, bias, Alog, dtA, dtB);
    cumsum_wmma_kernel<<<NH * 2, 512, 0, stream>>>(dtA, dtB, cw, ch);
    mask_write_kernel<<<NH * HH * WW, 256, 0, stream>>>(cw, ch, out);
}